// Top_Left_Corner_66623532695960
// MI455X (gfx1250) — compile-verified
//
#include <hip/hip_runtime.h>
#include <cstdint>
#include <cstddef>

#define C_DIM  256
#define H_DIM  512
#define W_DIM  512
#define NWAVE  16                  // waves per block (512 threads, wave32)
#define K_ROWS 8                   // rows per tile
#define NTILES (H_DIM / K_ROWS)    // 64

// out[c,h,w] = 2 * max_{h'>=h, w'>=w} x[c,h',w']
// One block per channel. Backward sweep over 8-row tiles:
//   rowsuffix_h[w] = reverse-cummax_w(x[h,:])   (independent of acc -> batched)
//   acc[w] = max(acc[w], rowsuffix_h[w]);  out[h,w] = 2*acc[w]
__global__ __launch_bounds__(512, 1) void tl_corner_pool_kernel(
    const float* __restrict__ x, float* __restrict__ out) {
  __shared__ __attribute__((aligned(16))) float tile[2][K_ROWS * W_DIM];  // 32 KB
  __shared__ float wtot[K_ROWS][NWAVE];
  __shared__ float wsuf[K_ROWS][NWAVE + 1];

  const int c    = blockIdx.x;
  const int w    = threadIdx.x;
  const int lane = w & 31;
  const int wid  = w >> 5;

  const float* xc = x   + (size_t)c * (H_DIM * W_DIM);
  float*       oc = out + (size_t)c * (H_DIM * W_DIM);

  if (w < K_ROWS) wsuf[w][NWAVE] = -__builtin_inff();  // sentinel above top wave

  // Workgroup-relative LDS byte addresses (low 32 bits of flat AS(3) address).
  const unsigned lds_slot[2] = {(unsigned)(size_t)(&tile[0][0]),
                                (unsigned)(size_t)(&tile[1][0])};

  // Tile T = global rows [H - K*(T+1), H-1-K*T]: one contiguous 16 KB block.
  // 512 threads x 2 async b128 loads cover it; slot = T & 1.
  auto issue_tile = [&](int T) {
    const unsigned slotbase = lds_slot[T & 1];
    const uint64_t gbase =
        (uint64_t)(size_t)(xc + (size_t)(H_DIM - K_ROWS * (T + 1)) * W_DIM);
#pragma unroll
    for (int j = 0; j < 2; ++j) {
      const unsigned off = (unsigned)((j * 512 + w) * 16);
      const unsigned la  = slotbase + off;
      const uint64_t ga  = gbase + off;
      asm volatile("global_load_async_to_lds_b128 %0, %1, off"
                   :: "v"(la), "v"(ga) : "memory");
    }
  };

  // Prologue: double-buffer tiles 0 and 1; make tile 0 resident everywhere.
  issue_tile(0);
  issue_tile(1);
  asm volatile("s_wait_asynccnt 0x2" ::: "memory");  // tile 0's 2 instrs retired
  __syncthreads();

  float acc = -__builtin_inff();

  for (int T = 0; T < NTILES; ++T) {
    const float* lslot = &tile[T & 1][0];

    // Row h_k = H-1 - (T*K + k)  ->  tile-relative row index K-1-k.
    float s[K_ROWS];
#pragma unroll
    for (int k = 0; k < K_ROWS; ++k)
      s[k] = lslot[(K_ROWS - 1 - k) * W_DIM + w];

    // 8 independent intra-wave reverse inclusive max-scans (co-issuing ILP).
#pragma unroll
    for (int off = 1; off < 32; off <<= 1) {
#pragma unroll
      for (int k = 0; k < K_ROWS; ++k) {
        float o = __shfl_down(s[k], off, 32);
        s[k] = (lane + off < 32) ? fmaxf(s[k], o) : s[k];
      }
    }

    if (lane == 0) {
#pragma unroll
      for (int k = 0; k < K_ROWS; ++k) wtot[k][wid] = s[k];  // wave totals
    }
    __syncthreads();  // barrier B: all tile reads + wave totals done

    if (wid == 0) {
      // Cross-wave suffix over 16 wave totals, 2 rows per pass via
      // width-16 segmented shuffles (lanes 0-15 -> row 2p, 16-31 -> row 2p+1).
#pragma unroll
      for (int p = 0; p < K_ROWS / 2; ++p) {
        const int kk  = 2 * p + (lane >> 4);
        const int seg = lane & 15;
        float t = wtot[kk][seg];
#pragma unroll
        for (int off = 1; off < 16; off <<= 1) {
          float o = __shfl_down(t, off, 16);
          t = (seg + off < 16) ? fmaxf(t, o) : t;
        }
        wsuf[kk][seg] = t;  // wsuf[k][m] = max over waves >= m
      }
    }

    // Refill the just-consumed slot with tile T+2 (all reads done at barrier B;
    // drain own DS ops first: async LDS writes are unordered vs DS).
    if (T + 2 < NTILES) {
      asm volatile("s_wait_dscnt 0x0" ::: "memory");
      issue_tile(T + 2);
      asm volatile("s_wait_asynccnt 0x2" ::: "memory");  // tile T+1 landed (in-order)
    } else {
      asm volatile("s_wait_asynccnt 0x0" ::: "memory");  // drain tail
    }
    __syncthreads();  // barrier C: wsuf ready AND tile T+1 resident everywhere

    // Serial fold across the 8 rows (descending h) + streaming NT stores.
    const size_t hbase = (size_t)(H_DIM - 1 - T * K_ROWS);
#pragma unroll
    for (int k = 0; k < K_ROWS; ++k) {
      const float rs = fmaxf(s[k], wsuf[k][wid + 1]);  // full row suffix at w
      acc = fmaxf(acc, rs);
      __builtin_nontemporal_store(acc + acc, &oc[(hbase - k) * W_DIM + w]);
    }
  }
}

extern "C" void kernel_launch(void* const* d_in, const int* in_sizes, int n_in,
                              void* d_out, int out_size, void* d_ws, size_t ws_size,
                              hipStream_t stream) {
  (void)in_sizes; (void)n_in; (void)out_size; (void)d_ws; (void)ws_size;
  const float* x = (const float*)d_in[0];
  float* out = (float*)d_out;
  tl_corner_pool_kernel<<<dim3(C_DIM), dim3(W_DIM), 0, stream>>>(x, out);
}